// Net_68599217651837
// MI455X (gfx1250) — compile-verified
//
#include <hip/hip_runtime.h>
#include <math.h>

typedef __attribute__((ext_vector_type(2))) float v2f;
typedef __attribute__((ext_vector_type(8))) float v8f;

#define BN_EPS 1e-5f
#define LSTR 68   // LDS row stride (floats): 4m+c bank pattern -> conflict-free b64 frags

// ---------------- init: deg = 1 (self loops), stats = 0 ----------------
__global__ __launch_bounds__(256) void k_init(float* __restrict__ deg,
                                              float* __restrict__ stats, int E) {
  int i = blockIdx.x * 256 + threadIdx.x;
  if (i < E) deg[i] = 1.0f;
  if (i < 256) stats[i] = 0.0f;
}

// ---------------- degree: deg[dst] += 1 per edge ----------------
__global__ __launch_bounds__(256) void k_deg(const int* __restrict__ eidx,
                                             float* __restrict__ deg, int E) {
  int e = blockIdx.x * 256 + threadIdx.x;
  if (e < E) atomicAdd(&deg[eidx[E + e]], 1.0f);
}

// ---------------- GEMM1: h1[E,128] = concat(x[src],x[dst]) @ W1^T ----------------
// block = 128 threads (4 waves); block tile = 64 edges x 128 outputs; K=128 in 2 halves.
__global__ __launch_bounds__(128) void k_gemm1(const float* __restrict__ x,
                                               const int* __restrict__ eidx,
                                               const float* __restrict__ W1,
                                               float* __restrict__ h1, int E) {
  __shared__ float Ws[128 * LSTR]; // W1[n][k-half], padded
  __shared__ float Xs[64 * LSTR];  // gathered rows, padded

  const int t = threadIdx.x;
  const int lane = t & 31;
  const int w = t >> 5;
  const int e0 = blockIdx.x * 64;

  v8f acc[8];
#pragma unroll
  for (int i = 0; i < 8; ++i)
#pragma unroll
    for (int j = 0; j < 8; ++j) acc[i][j] = 0.0f;

  const int mA = w * 16 + (lane & 15);
  const int koff = (lane >> 4) * 2;
  const int nL = lane & 15;

  for (int kh = 0; kh < 2; ++kh) {
    // stage W half: 128 rows x 64 cols (pairs)
    for (int p = t; p < 4096; p += 128) {
      int n = p >> 5, kk = (p & 31) * 2;
      const float* g = W1 + n * 128 + kh * 64 + kk;
      Ws[n * LSTR + kk] = g[0];
      Ws[n * LSTR + kk + 1] = g[1];
    }
    // stage gathered x rows: khalf0 -> x[src], khalf1 -> x[dst]
    for (int p = t; p < 2048; p += 128) {
      int m = p >> 5, kk = (p & 31) * 2;
      int node = eidx[kh * E + e0 + m];
      const float* g = x + node * 64 + kk;
      Xs[m * LSTR + kk] = g[0];
      Xs[m * LSTR + kk + 1] = g[1];
    }
    __syncthreads();
#pragma unroll
    for (int ks = 0; ks < 16; ++ks) {
      const int k4 = ks * 4;
      v2f a = *(const v2f*)&Xs[mA * LSTR + k4 + koff];
#pragma unroll
      for (int nt = 0; nt < 8; ++nt) {
        v2f b = *(const v2f*)&Ws[(nt * 16 + nL) * LSTR + k4 + koff];
        acc[nt] = __builtin_amdgcn_wmma_f32_16x16x4_f32(
            false, a, false, b, (short)0, acc[nt], false, false);
      }
    }
    __syncthreads();
  }

  const int mrow = e0 + w * 16 + ((lane >> 4) * 8);
  const int ncol = lane & 15;
#pragma unroll
  for (int nt = 0; nt < 8; ++nt)
#pragma unroll
    for (int v = 0; v < 8; ++v)
      h1[(mrow + v) * 128 + nt * 16 + ncol] = acc[nt][v];
}

// ---------------- self-loop + bias init: out = h/deg + b ----------------
__global__ __launch_bounds__(256) void k_selfbias(const float* __restrict__ h,
                                                  const float* __restrict__ deg,
                                                  const float* __restrict__ bias,
                                                  float* __restrict__ out,
                                                  int total4, int fshift) {
  int i = blockIdx.x * 256 + threadIdx.x;
  if (i >= total4) return;
  int row = i >> fshift;
  int fc = (i & ((1 << fshift) - 1)) << 2;
  int F = 4 << fshift;
  float inv = 1.0f / deg[row];
  float4 hv = *(const float4*)(h + row * F + fc);
  float4 bv = *(const float4*)(bias + fc);
  float4 o;
  o.x = hv.x * inv + bv.x;
  o.y = hv.y * inv + bv.y;
  o.z = hv.z * inv + bv.z;
  o.w = hv.w * inv + bv.w;
  *(float4*)(out + row * F + fc) = o;
}

// ---------------- edge scatter: out[dst] += norm * h[src] (f32 atomics) ----------------
__global__ __launch_bounds__(256) void k_edge(const int* __restrict__ eidx,
                                              const float* __restrict__ deg,
                                              const float* __restrict__ h,
                                              float* __restrict__ out,
                                              int E, int eshift) {
  int tid = blockIdx.x * 256 + threadIdx.x;
  int e = tid >> eshift;
  if (e >= E) return;
  int fc = (tid & ((1 << eshift) - 1)) << 2;
  int F = 4 << eshift;
  int s = eidx[e];
  int d = eidx[E + e];
  float nrm = rsqrtf(deg[s]) * rsqrtf(deg[d]);
  float4 hv = *(const float4*)(h + s * F + fc);
  float* o = out + d * F + fc;
  atomicAdd(o + 0, nrm * hv.x);
  atomicAdd(o + 1, nrm * hv.y);
  atomicAdd(o + 2, nrm * hv.z);
  atomicAdd(o + 3, nrm * hv.w);
}

// ---------------- BN stats: per-feature sum / sumsq ----------------
__global__ __launch_bounds__(128) void k_stats(const float* __restrict__ h,
                                               float* __restrict__ stats, int E) {
  int f = threadIdx.x; // 0..127
  float s = 0.0f, sq = 0.0f;
  for (int r = blockIdx.x; r < E; r += gridDim.x) {
    float v = h[r * 128 + f];
    s += v;
    sq += v * v;
  }
  atomicAdd(&stats[f], s);
  atomicAdd(&stats[128 + f], sq);
}

// ---------------- BN apply + ReLU ----------------
__global__ __launch_bounds__(256) void k_bnrelu(const float* __restrict__ h,
                                                const float* __restrict__ stats,
                                                const float* __restrict__ gamma,
                                                const float* __restrict__ beta,
                                                float* __restrict__ out, int E) {
  int i = blockIdx.x * 256 + threadIdx.x; // float4 index over E*128
  int total4 = E * 32;
  if (i >= total4) return;
  int fc = (i & 31) * 4;
  float invE = 1.0f / (float)E;
  float4 hv = *(const float4*)(h + i * 4);
  float4 sm = *(const float4*)(stats + fc);
  float4 sq = *(const float4*)(stats + 128 + fc);
  float4 g = *(const float4*)(gamma + fc);
  float4 bt = *(const float4*)(beta + fc);
  float4 o;
#define BN1(c)                                                  \
  {                                                             \
    float mu = sm.c * invE;                                     \
    float var = sq.c * invE - mu * mu;                          \
    float sc = g.c * rsqrtf(var + BN_EPS);                      \
    float y = (hv.c - mu) * sc + bt.c;                          \
    o.c = y > 0.0f ? y : 0.0f;                                  \
  }
  BN1(x) BN1(y) BN1(z) BN1(w)
#undef BN1
  *(float4*)(out + i * 4) = o;
}

// ---------------- GEMM2: h2[E,64] = hin[E,128] @ W2^T ----------------
__global__ __launch_bounds__(128) void k_gemm2(const float* __restrict__ hin,
                                               const float* __restrict__ W2,
                                               float* __restrict__ h2, int E) {
  __shared__ float Ws[64 * LSTR];
  __shared__ float Xs[64 * LSTR];
  const int t = threadIdx.x;
  const int lane = t & 31;
  const int w = t >> 5;
  const int e0 = blockIdx.x * 64;

  v8f acc[4];
#pragma unroll
  for (int i = 0; i < 4; ++i)
#pragma unroll
    for (int j = 0; j < 8; ++j) acc[i][j] = 0.0f;

  const int mA = w * 16 + (lane & 15);
  const int koff = (lane >> 4) * 2;
  const int nL = lane & 15;

  for (int kh = 0; kh < 2; ++kh) {
    for (int p = t; p < 2048; p += 128) { // 64 x 64 W half
      int n = p >> 5, kk = (p & 31) * 2;
      const float* g = W2 + n * 128 + kh * 64 + kk;
      Ws[n * LSTR + kk] = g[0];
      Ws[n * LSTR + kk + 1] = g[1];
    }
    for (int p = t; p < 2048; p += 128) { // 64 rows x 64 cols of hin
      int m = p >> 5, kk = (p & 31) * 2;
      const float* g = hin + (e0 + m) * 128 + kh * 64 + kk;
      Xs[m * LSTR + kk] = g[0];
      Xs[m * LSTR + kk + 1] = g[1];
    }
    __syncthreads();
#pragma unroll
    for (int ks = 0; ks < 16; ++ks) {
      const int k4 = ks * 4;
      v2f a = *(const v2f*)&Xs[mA * LSTR + k4 + koff];
#pragma unroll
      for (int nt = 0; nt < 4; ++nt) {
        v2f b = *(const v2f*)&Ws[(nt * 16 + nL) * LSTR + k4 + koff];
        acc[nt] = __builtin_amdgcn_wmma_f32_16x16x4_f32(
            false, a, false, b, (short)0, acc[nt], false, false);
      }
    }
    __syncthreads();
  }

  const int mrow = e0 + w * 16 + ((lane >> 4) * 8);
  const int ncol = lane & 15;
#pragma unroll
  for (int nt = 0; nt < 4; ++nt)
#pragma unroll
    for (int v = 0; v < 8; ++v)
      h2[(mrow + v) * 64 + nt * 16 + ncol] = acc[nt][v];
}

// ---------------- final: p = sigmoid(h @ Wl^T + bl); rounded = p>=cutoff ----------------
__global__ __launch_bounds__(128) void k_final(const float* __restrict__ hin,
                                               const float* __restrict__ Wlg,
                                               const float* __restrict__ bl,
                                               const float* __restrict__ cutp,
                                               float* __restrict__ outp, int E) {
  __shared__ float Ws[64 * LSTR];
  __shared__ float Xs[64 * LSTR];
  const int t = threadIdx.x;
  const int lane = t & 31;
  const int w = t >> 5;
  const int e0 = blockIdx.x * 64;

  v8f acc[4];
#pragma unroll
  for (int i = 0; i < 4; ++i)
#pragma unroll
    for (int j = 0; j < 8; ++j) acc[i][j] = 0.0f;

  for (int p = t; p < 2048; p += 128) { // full 64x64 Wl
    int n = p >> 5, kk = (p & 31) * 2;
    const float* g = Wlg + n * 64 + kk;
    Ws[n * LSTR + kk] = g[0];
    Ws[n * LSTR + kk + 1] = g[1];
  }
  for (int p = t; p < 2048; p += 128) { // 64 rows x 64 cols of hin
    int m = p >> 5, kk = (p & 31) * 2;
    const float* g = hin + (e0 + m) * 64 + kk;
    Xs[m * LSTR + kk] = g[0];
    Xs[m * LSTR + kk + 1] = g[1];
  }
  __syncthreads();

  const int mA = w * 16 + (lane & 15);
  const int koff = (lane >> 4) * 2;
  const int nL = lane & 15;
#pragma unroll
  for (int ks = 0; ks < 16; ++ks) {
    const int k4 = ks * 4;
    v2f a = *(const v2f*)&Xs[mA * LSTR + k4 + koff];
#pragma unroll
    for (int nt = 0; nt < 4; ++nt) {
      v2f b = *(const v2f*)&Ws[(nt * 16 + nL) * LSTR + k4 + koff];
      acc[nt] = __builtin_amdgcn_wmma_f32_16x16x4_f32(
          false, a, false, b, (short)0, acc[nt], false, false);
    }
  }

  const float cut = *cutp;
  const int mrow = e0 + w * 16 + ((lane >> 4) * 8);
  const int ncol = lane & 15;
#pragma unroll
  for (int nt = 0; nt < 4; ++nt) {
    const int col = nt * 16 + ncol;
    const float bv = bl[col];
#pragma unroll
    for (int v = 0; v < 8; ++v) {
      float xv = acc[nt][v] + bv;
      float pv = 1.0f / (1.0f + __expf(-xv));
      outp[(mrow + v) * 64 + col] = pv;
      outp[E * 64 + (mrow + v) * 64 + col] = (pv < cut) ? 0.0f : 1.0f;
    }
  }
}

extern "C" void kernel_launch(void* const* d_in, const int* in_sizes, int n_in,
                              void* d_out, int out_size, void* d_ws, size_t ws_size,
                              hipStream_t stream) {
  const float* x     = (const float*)d_in[0];
  const int*   eidx  = (const int*)d_in[1];
  const float* W1    = (const float*)d_in[2];
  const float* b1    = (const float*)d_in[3];
  const float* gamma = (const float*)d_in[4];
  const float* beta  = (const float*)d_in[5];
  const float* W2    = (const float*)d_in[6];
  const float* b2    = (const float*)d_in[7];
  const float* Wl    = (const float*)d_in[8];
  const float* bl    = (const float*)d_in[9];
  const float* cut   = (const float*)d_in[10];
  float* out = (float*)d_out;

  const int E = in_sizes[1] / 2; // 640000, divisible by 64

  // workspace layout (floats)
  float* bufA  = (float*)d_ws;          // E*128
  float* bufB  = bufA + (size_t)E * 128; // E*128
  float* deg   = bufB + (size_t)E * 128; // E
  float* stats = deg + E;                // 256

  const int eb = (E + 255) / 256;
  k_init<<<eb, 256, 0, stream>>>(deg, stats, E);
  k_deg<<<eb, 256, 0, stream>>>(eidx, deg, E);

  // Stage 1: GEMM1 -> bufA, agg(self+bias then edge atomics) -> bufB
  k_gemm1<<<E / 64, 128, 0, stream>>>(x, eidx, W1, bufA, E);
  k_selfbias<<<(E * 32 + 255) / 256, 256, 0, stream>>>(bufA, deg, b1, bufB, E * 32, 5);
  k_edge<<<(E * 32 + 255) / 256, 256, 0, stream>>>(eidx, deg, bufA, bufB, E, 5);

  // BatchNorm (training stats) + ReLU: bufB -> bufA
  k_stats<<<1024, 128, 0, stream>>>(bufB, stats, E);
  k_bnrelu<<<(E * 32 + 255) / 256, 256, 0, stream>>>(bufB, stats, gamma, beta, bufA, E);

  // Stage 2: GEMM2 bufA -> bufB[E*64], agg -> bufA[E*64]
  k_gemm2<<<E / 64, 128, 0, stream>>>(bufA, W2, bufB, E);
  k_selfbias<<<(E * 16 + 255) / 256, 256, 0, stream>>>(bufB, deg, b2, bufA, E * 16, 4);
  k_edge<<<(E * 16 + 255) / 256, 256, 0, stream>>>(eidx, deg, bufB, bufA, E, 4);

  // Stage 3: final linear + sigmoid + cutoff -> d_out (p then rounded)
  k_final<<<E / 64, 128, 0, stream>>>(bufA, Wl, bl, cut, out, E);
}